// AutoReconstruction_79104707658019
// MI455X (gfx1250) — compile-verified
//
#include <hip/hip_runtime.h>

// out[b,m] = dot(inputs[b,m,:], W[m,:]) + bias[m]
// B=1024, M=2048, I=128, all fp32.
//
// Per 16-wide m-tile: out[b, m0:m0+16] = diag( X_tile(16x128) * W_tile(16x128)^T )
// computed with V_WMMA_F32_16X16X4_F32 accumulated over K=128 (32 steps).
//
// K-column remap for b128 loads: steps {2s, 2s+1} consume columns {8s..8s+7}:
//   step 2s  : lo-lanes K-slots = cols {8s+0, 8s+1}, hi-lanes = {8s+4, 8s+5}
//   step 2s+1: lo-lanes K-slots = cols {8s+2, 8s+3}, hi-lanes = {8s+6, 8s+7}
// A and B use the identical mapping, so the K-sum is unchanged.

typedef float v2f __attribute__((ext_vector_type(2)));
typedef float v4f __attribute__((ext_vector_type(4)));
typedef float v8f __attribute__((ext_vector_type(8)));

constexpr int Bdim = 1024;
constexpr int Mdim = 2048;
constexpr int Idim = 128;

constexpr int WAVE_SIZE       = 32;
constexpr int WAVES_PER_BLOCK = 4;                 // 128 threads/block
constexpr int MTILE           = 16;                // one WMMA tile of m per wave
constexpr int B_CHUNKS        = 64;                // grid.y
constexpr int B_PER_BLOCK     = Bdim / B_CHUNKS;   // 16 batches per block

__global__ __launch_bounds__(WAVES_PER_BLOCK * WAVE_SIZE)
void diag_linear_wmma_f32(const float* __restrict__ x,     // [B, M, I]
                          const float* __restrict__ w,     // [M, I]
                          const float* __restrict__ bias,  // [M]
                          float* __restrict__ out)         // [B, M]
{
    const int lane   = threadIdx.x & (WAVE_SIZE - 1);
    const int waveId = threadIdx.x >> 5;

    const int m0    = (blockIdx.x * WAVES_PER_BLOCK + waveId) * MTILE;
    const int bBase = blockIdx.y * B_PER_BLOCK;
    if (m0 >= Mdim || bBase >= Bdim) return;   // wave-uniform guard (EXEC stays full)

    // A-matrix 16x4 fp32 layout (ISA 7.12.2):
    //   VGPR0: lanes 0-15 -> K slot 0, lanes 16-31 -> K slot 2
    //   VGPR1: lanes 0-15 -> K slot 1, lanes 16-31 -> K slot 3
    // B-matrix 4x16 mirrors it with lanes holding N.
    const int row    = lane & 15;   // M row for A == N column for B
    const int laneHi = lane >> 4;   // 0 -> lo K slots, 1 -> hi K slots

    // ---- Hoist B operand (W_tile^T) into registers: reused for every b ----
    // One b128 per lane feeds two consecutive WMMA steps.
    const float* wRow = w + (size_t)(m0 + row) * Idim + 4 * laneHi;
    v2f bop[Idim / 4];
#pragma unroll
    for (int s = 0; s < Idim / 8; ++s) {
        v4f t = *(const v4f*)(wRow + 8 * s);
        bop[2 * s]     = __builtin_shufflevector(t, t, 0, 1);
        bop[2 * s + 1] = __builtin_shufflevector(t, t, 2, 3);
    }

    // Per-lane output column for the diagonal write-back.
    //   diag(v,v)     lives in C[v] at lane v        (v = 0..7)
    //   diag(8+v,8+v) lives in C[v] at lane 24+v
    const int  j      = (lane < 16) ? lane : (lane - 16);   // 0..15, always valid
    const bool active = (lane < 8) || (lane >= 24);
    const float bj    = bias[m0 + j];

    for (int bi = 0; bi < B_PER_BLOCK; ++bi) {
        const int b = bBase + bi;
        const float* xRow =
            x + ((size_t)b * Mdim + (m0 + row)) * Idim + 4 * laneHi;

        // Pre-touch next batch's row: lowers to global_prefetch_b8 on gfx1250.
        if (bi + 1 < B_PER_BLOCK)
            __builtin_prefetch(xRow + (size_t)Mdim * Idim, 0, 0);

        v8f c = {0.f, 0.f, 0.f, 0.f, 0.f, 0.f, 0.f, 0.f};
#pragma unroll
        for (int s = 0; s < Idim / 8; ++s) {
            // One contiguous 16B load per lane -> two WMMA K-steps.
            v4f a = *(const v4f*)(xRow + 8 * s);
            v2f a01 = __builtin_shufflevector(a, a, 0, 1);
            v2f a23 = __builtin_shufflevector(a, a, 2, 3);
            // 8 args: (neg_a, A, neg_b, B, c_mod, C, reuse_a, reuse_b)
            c = __builtin_amdgcn_wmma_f32_16x16x4_f32(
                    false, a01, false, bop[2 * s], (short)0, c, false, false);
            c = __builtin_amdgcn_wmma_f32_16x16x4_f32(
                    false, a23, false, bop[2 * s + 1], (short)0, c, false, false);
        }

        // Branch-free diagonal extraction from the C/D layout.
        float diag = c[0];
#pragma unroll
        for (int v = 1; v < 8; ++v)
            if ((j & 7) == v) diag = c[v];

        if (active)
            out[(size_t)b * Mdim + m0 + j] = diag + bj;
    }
}

extern "C" void kernel_launch(void* const* d_in, const int* in_sizes, int n_in,
                              void* d_out, int out_size, void* d_ws, size_t ws_size,
                              hipStream_t stream) {
    const float* x    = (const float*)d_in[0];   // inputs   [B, M, I]
    const float* wgt  = (const float*)d_in[1];   // Rk_weight[M, I]
    const float* bias = (const float*)d_in[2];   // bias     [M]
    float* out        = (float*)d_out;           // [B, M]

    dim3 grid(Mdim / (WAVES_PER_BLOCK * MTILE),  // 32 m-tile groups
              B_CHUNKS);                         // 64 b-chunks
    dim3 block(WAVES_PER_BLOCK * WAVE_SIZE);     // 128 threads = 4 waves

    diag_linear_wmma_f32<<<grid, block, 0, stream>>>(x, wgt, bias, out);
}